// LossWrapper_33174327394875
// MI455X (gfx1250) — compile-verified
//
#include <hip/hip_runtime.h>
#include <hip/hip_bf16.h>
#include <math.h>

// ---------------------------------------------------------------------------
// Problem constants (from the reference)
// ---------------------------------------------------------------------------
#define M_ROWS   32
#define S_ROWS   32
#define FEAT     192
#define NCLASS   100000
#define S_SCALE  30.0f
#define NCHUNK   98          // ceil(100000 / 1024)
#define CHUNK    1024

typedef float v2f  __attribute__((ext_vector_type(2)));
typedef float v8f  __attribute__((ext_vector_type(8)));
typedef unsigned int u32x4 __attribute__((ext_vector_type(4)));
typedef int   i32x4 __attribute__((ext_vector_type(4)));
typedef int   i32x8 __attribute__((ext_vector_type(8)));

// Workspace layout (floats):
//  ws_xn    : 32*192            = 6144
//  ws_stats : 32*8              = 256    (cmi, smi, th, mm, reg, ...)
//  ws_cp    : 100000*64         = 6,400,000   ([n][2m]=cos, [n][2m+1]=phi)
//  ws_part  : 1024*98*2         = 200,704     ((max,sum) per (pair,chunk))
#define WS_XN       0
#define WS_STATS    6144
#define WS_CP_OFF   (6144 + 256)
#define WS_PART_OFF (6144 + 256 + NCLASS * 64)

// ---------------------------------------------------------------------------
// Kernel 0: per-row embedding stats + normalized embeddings
// ---------------------------------------------------------------------------
__global__ void k_stats(const float* __restrict__ pred_embs,
                        float* __restrict__ ws) {
    int m = threadIdx.x;
    if (m >= M_ROWS) return;
    float* ws_xn    = ws + WS_XN;
    float* ws_stats = ws + WS_STATS;

    float ss = 0.0f;
    for (int k = 0; k < FEAT; ++k) {
        float e = pred_embs[m * FEAT + k];
        ss += e * e;
    }
    float nrm  = sqrtf(ss);
    float ninv = 1.0f / fmaxf(nrm, 1e-12f);
    for (int k = 0; k < FEAT; ++k)
        ws_xn[m * FEAT + k] = pred_embs[m * FEAT + k] * ninv;

    float nc  = fminf(fmaxf(nrm, 10.0f), 110.0f);          // clip(norm, LOW, HIGH)
    float mi  = 0.01f * nc + 100.0f / nc;                  // ALPHA*n + BETA/n
    float cmi = cosf(mi);
    float smi = sinf(mi);
    ws_stats[m * 8 + 0] = cmi;
    ws_stats[m * 8 + 1] = smi;
    ws_stats[m * 8 + 2] = -cmi;            // th = cos(pi - mi)
    ws_stats[m * 8 + 3] = smi * mi;        // mm = sin(pi - mi) * mi
    ws_stats[m * 8 + 4] = (nc - 60.0f) * (nc - 60.0f);   // reg
}

// ---------------------------------------------------------------------------
// Kernel 1: WMMA f32 GEMM (xn . wn^T) fused with MagFace phi.
// The 24KB xn tile is staged into LDS by the Tensor Data Mover (one
// tensor_load_to_lds by wave 0, s_wait_tensorcnt, barrier); each wave then
// computes a 32(m) x 16(n) tile via two v_wmma_f32_16x16x4_f32 accumulators
// over K=192 (48 steps), normalizing weight rows on the fly.
// ---------------------------------------------------------------------------
__global__ void k_gemm(const float* __restrict__ weight,
                       float* __restrict__ ws) {
    const float* ws_xn    = ws + WS_XN;
    const float* ws_stats = ws + WS_STATS;
    float*       ws_cp    = ws + WS_CP_OFF;

    __shared__ float s_xn[M_ROWS * FEAT];    // 24 KB
    __shared__ float s_st[M_ROWS * 4];       // cmi, smi, th, mm

    int tid = threadIdx.x;
    if (tid < M_ROWS * 4) s_st[tid] = ws_stats[(tid >> 2) * 8 + (tid & 3)];

#if __has_builtin(__builtin_amdgcn_tensor_load_to_lds)
    if (tid < 32) {
        // --- Tensor DMA descriptor (ISA §8.3/8.4): 2-D 192x32 f32 tile ---
        unsigned long long ga = (unsigned long long)(const void*)ws_xn;
        unsigned int lds_off  = (unsigned int)(unsigned long long)(void*)s_xn;
        u32x4 g0;
        g0.x = 1u;                                       // count=1, user mode
        g0.y = lds_off;                                  // lds_addr
        g0.z = (unsigned int)(ga & 0xFFFFFFFFu);         // global_addr[31:0]
        g0.w = (unsigned int)((ga >> 32) & 0x1FFFFFFu)   // global_addr[56:32]
             | (2u << 30);                               // type = 2 (image)
        i32x8 g1;
        g1[0] = (2 << 16);            // data_size = 4B
        g1[1] = (FEAT << 16);         // tensor_dim0[15:0] = 192
        g1[2] = (M_ROWS << 16);       // tensor_dim0 hi=0 | tensor_dim1[15:0]=32
        g1[3] = (FEAT << 16);         // tensor_dim1 hi=0 | tile_dim0=192
        g1[4] = M_ROWS;               // tile_dim1=32 | tile_dim2=0
        g1[5] = FEAT;                 // tensor_dim0_stride[31:0] = 192
        g1[6] = 0;                    // stride hi | tensor_dim1_stride lo
        g1[7] = 0;
        i32x4 g2 = {0, 0, 0, 0};
        i32x4 g3 = {0, 0, 0, 0};
        i32x8 g4 = {0, 0, 0, 0, 0, 0, 0, 0};
        __builtin_amdgcn_tensor_load_to_lds(g0, g1, g2, g3, g4, 0);
        __builtin_amdgcn_s_wait_tensorcnt(0);
    }
#else
    for (int i = tid; i < M_ROWS * FEAT; i += 256) s_xn[i] = ws_xn[i];
#endif
    __syncthreads();

    int wave = tid >> 5;          // 0..7
    int lane = tid & 31;
    int col  = lane & 15;         // N within tile, also A row (M % 16)
    int kh   = lane >> 4;         // K-half select per ISA fragment layout

    long n  = (long)blockIdx.x * 128 + wave * 16 + col;
    long ncl = (n < NCLASS) ? n : (NCLASS - 1);     // clamp OOB loads
    const float* wrow = weight + ncl * FEAT;

    // ---- weight row norm (each half-wave sums half of K, combine with xor16)
    float ss = 0.0f;
    for (int k = kh * 96; k < kh * 96 + 96; ++k) {
        float w = wrow[k];
        ss += w * w;
    }
    ss += __shfl_xor(ss, 16, 32);
    float rinv = 1.0f / fmaxf(sqrtf(ss), 1e-12f);

    // ---- WMMA accumulation: A(16x4 f32) x B(4x16 f32) + C(16x16 f32)
    v8f c0 = {};   // rows m = 0..15
    v8f c1 = {};   // rows m = 16..31
    for (int k0 = 0; k0 < FEAT; k0 += 4) {
        int k = k0 + 2 * kh;      // per-lane K per fragment layout (§7.12.2)
        v2f a0, a1, b;
        a0.x = s_xn[col * FEAT + k];
        a0.y = s_xn[col * FEAT + k + 1];
        a1.x = s_xn[(col + 16) * FEAT + k];
        a1.y = s_xn[(col + 16) * FEAT + k + 1];
        b.x  = wrow[k] * rinv;
        b.y  = wrow[k + 1] * rinv;
        c0 = __builtin_amdgcn_wmma_f32_16x16x4_f32(false, a0, false, b,
                                                   (short)0, c0, false, false);
        c1 = __builtin_amdgcn_wmma_f32_16x16x4_f32(false, a1, false, b,
                                                   (short)0, c1, false, false);
    }

    // ---- fused phi + interleaved store ([n][2m]=cos, [n][2m+1]=phi)
    if (n < NCLASS) {
        float* outp = ws_cp + n * 64;
        int mbase = (lane < 16) ? 0 : 8;   // C layout: lanes 16-31 hold M+8
        float buf0[16], buf1[16];
        #pragma unroll
        for (int j = 0; j < 8; ++j) {
            {
                int m  = mbase + j;
                float cs   = c0[j];
                float sine = sqrtf(fmaxf(1.0f - cs * cs, 0.0f));
                float cmi = s_st[m * 4 + 0], smi = s_st[m * 4 + 1];
                float th  = s_st[m * 4 + 2], mm  = s_st[m * 4 + 3];
                float ph = (cs > th) ? (cs * cmi - sine * smi) : (cs - mm);
                buf0[2 * j]     = cs;
                buf0[2 * j + 1] = ph;
            }
            {
                int m  = mbase + j + 16;
                float cs   = c1[j];
                float sine = sqrtf(fmaxf(1.0f - cs * cs, 0.0f));
                float cmi = s_st[m * 4 + 0], smi = s_st[m * 4 + 1];
                float th  = s_st[m * 4 + 2], mm  = s_st[m * 4 + 3];
                float ph = (cs > th) ? (cs * cmi - sine * smi) : (cs - mm);
                buf1[2 * j]     = cs;
                buf1[2 * j + 1] = ph;
            }
        }
        float4* p0 = (float4*)(outp + 2 * mbase);          // 16B aligned
        float4* p1 = (float4*)(outp + 2 * (mbase + 16));
        #pragma unroll
        for (int q = 0; q < 4; ++q) {
            p0[q] = ((const float4*)buf0)[q];
            p1[q] = ((const float4*)buf1)[q];
        }
    }
}

// ---------------------------------------------------------------------------
// Kernel 2: online logsumexp partials. grid = (98 chunks, 4 pair-groups),
// 256 threads. Thread -> pair (s,m); each wave covers one s and all 32 m,
// so per-n loads are a single 256B-contiguous b64 per wave-row, with a
// global_prefetch 8 rows ahead (L2-resident stream).
// ---------------------------------------------------------------------------
__global__ void k_lse(const float* __restrict__ ws_cp,
                      float* __restrict__ ws_part,
                      const int* __restrict__ gt) {
    int t    = threadIdx.x;
    int pair = blockIdx.y * 256 + t;     // 0..1023
    int s    = pair >> 5;
    int m    = pair & 31;

    float lv = (float)gt[s];
    float a  = S_SCALE * lv;             // coefficient of (phi - cos)

    int nbeg = blockIdx.x * CHUNK;
    int nend = nbeg + CHUNK;
    if (nend > NCLASS) nend = NCLASS;

    const float2* cp2 = (const float2*)ws_cp;   // [n][m] -> (cos, phi)

    float mx = -3.4e38f, sm = 0.0f;
    for (int n = nbeg; n < nend; ++n) {
        if (n + 8 < nend)
            __builtin_prefetch(&cp2[(n + 8) * 32 + m], 0, 1);
        float2 v = cp2[n * 32 + m];
        float x  = S_SCALE * v.x + a * (v.y - v.x);  // 30*(lv*phi + (1-lv)*cos)
        float nm = fmaxf(mx, x);
        sm = sm * __expf(mx - nm) + __expf(x - nm);
        mx = nm;
    }
    ws_part[(pair * NCHUNK + blockIdx.x) * 2]     = mx;
    ws_part[(pair * NCHUNK + blockIdx.x) * 2 + 1] = sm;
}

// ---------------------------------------------------------------------------
// Kernel 3: combine partials -> cost matrix -> Hungarian -> final scalar.
// ---------------------------------------------------------------------------
__global__ void k_final(const float* __restrict__ ws,
                        const int* __restrict__ gt,
                        const float* __restrict__ pred_ps,
                        float* __restrict__ out) {
    const float* ws_cp    = ws + WS_CP_OFF;
    const float* ws_part  = ws + WS_PART_OFF;
    const float* ws_stats = ws + WS_STATS;

    __shared__ double Cmat[S_ROWS * M_ROWS];

    int p = threadIdx.x;       // 0..1023 == (s,m) pair
    int s = p >> 5;
    int m = p & 31;

    float MX = -3.4e38f;
    for (int ch = 0; ch < NCHUNK; ++ch)
        MX = fmaxf(MX, ws_part[(p * NCHUNK + ch) * 2]);
    float SUM = 0.0f;
    for (int ch = 0; ch < NCHUNK; ++ch)
        SUM += ws_part[(p * NCHUNK + ch) * 2 + 1] *
               __expf(ws_part[(p * NCHUNK + ch) * 2] - MX);
    float lse = MX + logf(SUM);

    int   g  = gt[s];
    float lv = (float)g;
    float cs = ws_cp[(long)g * 64 + 2 * m];
    float ph = ws_cp[(long)g * 64 + 2 * m + 1];
    float tgt = S_SCALE * cs + S_SCALE * lv * (ph - cs);
    float reg = ws_stats[m * 8 + 4];

    Cmat[s * 32 + m] = (double)(lse - tgt) + 0.07 * (double)reg; // GAMMA*LMBDA*0.01
    __syncthreads();

    if (p == 0) {
        // ---- exact augmenting-path Hungarian (double), n = 32 ----
        const int n = 32;
        const double INF = 1e300;
        double u[33], v[33];
        int    pj[33], way[33];
        for (int j = 0; j <= n; ++j) { u[j] = 0.0; v[j] = 0.0; pj[j] = 0; way[j] = 0; }
        for (int i = 1; i <= n; ++i) {
            pj[0] = i;
            int j0 = 0;
            double minv[33];
            bool   used[33];
            for (int j = 0; j <= n; ++j) { minv[j] = INF; used[j] = false; }
            while (true) {
                used[j0] = true;
                int i0 = pj[j0];
                double delta = INF;
                int j1 = -1;
                for (int j = 1; j <= n; ++j) {
                    if (!used[j]) {
                        double cur = Cmat[(i0 - 1) * 32 + (j - 1)] - u[i0] - v[j];
                        if (cur < minv[j]) { minv[j] = cur; way[j] = j0; }
                        if (minv[j] < delta) { delta = minv[j]; j1 = j; }
                    }
                }
                for (int j = 0; j <= n; ++j) {
                    if (used[j]) { u[pj[j]] += delta; v[j] -= delta; }
                    else         { minv[j] -= delta; }
                }
                j0 = j1;
                if (pj[j0] == 0) break;
            }
            while (j0) { int j1 = way[j0]; pj[j0] = pj[j1]; j0 = j1; }
        }
        int col_of_row[32];
        for (int j = 1; j <= n; ++j) col_of_row[pj[j] - 1] = j - 1;

        // M == S: every row matched, t_exist all ones, t_stop = 1.
        double lspk = 0.0;
        for (int r = 0; r < 32; ++r) lspk += Cmat[r * 32 + col_of_row[r]];
        lspk /= 32.0;

        double lexist = 0.0;
        for (int i = 0; i < 32; ++i) {
            double pp = (double)pred_ps[i];
            pp = fmin(fmax(pp, 1e-6), 1.0 - 1e-6);
            lexist += -log(pp);
        }
        lexist /= 32.0;

        double lstop = log(2.0);   // -(1 * log(P_STOP)), P_STOP = 0.5
        out[0] = (float)(lspk + 100.0 * lexist + 100.0 * lstop);
    }
}

// ---------------------------------------------------------------------------
// Host launcher
// ---------------------------------------------------------------------------
extern "C" void kernel_launch(void* const* d_in, const int* in_sizes, int n_in,
                              void* d_out, int out_size, void* d_ws, size_t ws_size,
                              hipStream_t stream) {
    const float* pred_embs = (const float*)d_in[0];
    const float* pred_ps   = (const float*)d_in[1];
    const int*   gt_labels = (const int*)d_in[2];
    const float* weight    = (const float*)d_in[3];
    float*       out       = (float*)d_out;
    float*       ws        = (float*)d_ws;

    // 1) per-row embedding stats
    k_stats<<<1, 32, 0, stream>>>(pred_embs, ws);

    // 2) WMMA GEMM + phi: 782 blocks x 8 waves x 16 cols covers 100000 rows
    k_gemm<<<(NCLASS + 127) / 128, 256, 0, stream>>>(weight, ws);

    // 3) logsumexp partials
    k_lse<<<dim3(NCHUNK, 4), 256, 0, stream>>>(ws + WS_CP_OFF,
                                               ws + WS_PART_OFF, gt_labels);

    // 4) combine + Hungarian + final scalar
    k_final<<<1, 1024, 0, stream>>>(ws, gt_labels, pred_ps, out);
}